// GAT_NodeWeightEncoder_78572131713261
// MI455X (gfx1250) — compile-verified
//
#include <hip/hip_runtime.h>
#include <hip/hip_bf16.h>

typedef float v2f __attribute__((ext_vector_type(2)));
typedef float v8f __attribute__((ext_vector_type(8)));

#define IN_DIM 128
#define HC     256   // H*C = 4*64
#define NHEAD  4
#define CPH    64
#define NEG_SLOPE 0.2f
#define BN_EPS 1e-5f
#define ENC_NEGINF 0x007FFFFFu

// ---------- order-preserving float<->uint map for atomic max ----------
__device__ __forceinline__ unsigned enc_f32(float f) {
  unsigned u = __float_as_uint(f);
  return (u & 0x80000000u) ? ~u : (u | 0x80000000u);
}
__device__ __forceinline__ float dec_f32(unsigned u) {
  unsigned b = (u & 0x80000000u) ? (u ^ 0x80000000u) : ~u;
  return __uint_as_float(b);
}

// ---------- init scratch each call (graph-replay safe) ----------
__global__ void k_init(float* __restrict__ acc, float* __restrict__ den,
                       unsigned* __restrict__ mbits, float* __restrict__ stats,
                       int nacc, int nnh) {
  int i = blockIdx.x * blockDim.x + threadIdx.x;
  if (i < nacc) acc[i] = 0.0f;
  if (i < nnh)  { den[i] = 0.0f; mbits[i] = ENC_NEGINF; }
  if (i < 2048) stats[i] = 0.0f;
}

// ---------- xp = x @ W via V_WMMA_F32_16X16X4_F32 ----------
// block = 128 threads (4 waves). Block handles 16 rows x 256 cols.
// Wave w covers cols [64w, 64w+64) as 4 accumulator tiles, K-loop of 32 steps.
__global__ __launch_bounds__(128) void k_gemm(const float* __restrict__ x,
                                              const float* __restrict__ W,
                                              float* __restrict__ xp, int Nn) {
  __shared__ float Atile[16 * IN_DIM];  // 8 KB
  const int rt = blockIdx.x;
  const int t  = threadIdx.x;
  const int row0 = rt * 16;

  // cooperative A-tile load (contiguous 2048 floats)
  if (row0 + 16 <= Nn) {
    const float4* s4 = (const float4*)(x + (size_t)row0 * IN_DIM);
    float4* d4 = (float4*)Atile;
#pragma unroll
    for (int i = 0; i < 4; ++i) d4[t + 128 * i] = s4[t + 128 * i];
  } else {
#pragma unroll
    for (int i = 0; i < 16; ++i) {
      int f = t * 16 + i;
      int r = f >> 7, c = f & 127;
      Atile[f] = (row0 + r < Nn) ? x[(size_t)(row0 + r) * IN_DIM + c] : 0.0f;
    }
  }
  __syncthreads();

  const int lane = t & 31;
  const int hi   = lane >> 4;   // 0/1 selects K pair
  const int lm   = lane & 15;   // row / col within fragment
  const int colbase = (t >> 5) * 64;

  v8f acc0 = {}, acc1 = {}, acc2 = {}, acc3 = {};
#pragma unroll 4
  for (int k = 0; k < 32; ++k) {
    const int kb = k * 4 + 2 * hi;
    v2f a; a.x = Atile[lm * IN_DIM + kb]; a.y = Atile[lm * IN_DIM + kb + 1];
    const float* Wr0 = W + (size_t)(kb    ) * HC + colbase + lm;
    const float* Wr1 = W + (size_t)(kb + 1) * HC + colbase + lm;
    if (k + 1 < 32) __builtin_prefetch(Wr0 + 4 * HC, 0, 1);  // global_prefetch_b8
    v2f b0, b1, b2, b3;
    b0.x = Wr0[0];  b0.y = Wr1[0];
    b1.x = Wr0[16]; b1.y = Wr1[16];
    b2.x = Wr0[32]; b2.y = Wr1[32];
    b3.x = Wr0[48]; b3.y = Wr1[48];
    acc0 = __builtin_amdgcn_wmma_f32_16x16x4_f32(false, a, false, b0, (short)0, acc0, false, false);
    acc1 = __builtin_amdgcn_wmma_f32_16x16x4_f32(false, a, false, b1, (short)0, acc1, false, false);
    acc2 = __builtin_amdgcn_wmma_f32_16x16x4_f32(false, a, false, b2, (short)0, acc2, false, false);
    acc3 = __builtin_amdgcn_wmma_f32_16x16x4_f32(false, a, false, b3, (short)0, acc3, false, false);
  }

#pragma unroll
  for (int r = 0; r < 8; ++r) {
    const int row = row0 + r + 8 * hi;
    if (row < Nn) {
      float* o = xp + (size_t)row * HC + colbase + lm;
      o[0] = acc0[r]; o[16] = acc1[r]; o[32] = acc2[r]; o[48] = acc3[r];
    }
  }
}

// ---------- per-node attention scalars a_src/a_dst ----------
__global__ void k_attn(const float* __restrict__ xp, const float* __restrict__ att_src,
                       const float* __restrict__ att_dst, float* __restrict__ a_src,
                       float* __restrict__ a_dst, int nnh) {
  int idx = blockIdx.x * blockDim.x + threadIdx.x;
  if (idx >= nnh) return;
  int n = idx >> 2, h = idx & 3;
  const float* v  = xp + (size_t)n * HC + h * CPH;
  const float* as = att_src + h * CPH;
  const float* ad = att_dst + h * CPH;
  float s = 0.0f, d = 0.0f;
#pragma unroll 8
  for (int c = 0; c < CPH; ++c) { float t = v[c]; s += t * as[c]; d += t * ad[c]; }
  a_src[idx] = s; a_dst[idx] = d;
}

__device__ __forceinline__ float leaky(float x) { return x >= 0.0f ? x : NEG_SLOPE * x; }

// ---------- edge pass 1: segment max via uint-encoded atomicMax ----------
__global__ void k_edge_max(const long long* __restrict__ ei, const float* __restrict__ a_src,
                           const float* __restrict__ a_dst, unsigned* __restrict__ mbits, int Ee) {
  int e = blockIdx.x * blockDim.x + threadIdx.x;
  if (e >= Ee) return;
  int s = (int)ei[e], d = (int)ei[Ee + e];
#pragma unroll
  for (int h = 0; h < NHEAD; ++h) {
    float l = leaky(a_src[s * 4 + h] + a_dst[d * 4 + h]);
    atomicMax(&mbits[d * 4 + h], enc_f32(l));
  }
}

__global__ void k_node_m(const unsigned* __restrict__ mbits, float* __restrict__ m, int nnh) {
  int i = blockIdx.x * blockDim.x + threadIdx.x;
  if (i >= nnh) return;
  unsigned u = mbits[i];
  m[i] = (u == ENC_NEGINF) ? 0.0f : dec_f32(u);   // empty segment -> 0
}

// ---------- edge pass 2: e = exp(l - m[dst]); denom += e ----------
__global__ void k_edge_exp(const long long* __restrict__ ei, const float* __restrict__ a_src,
                           const float* __restrict__ a_dst, const float* __restrict__ m,
                           float* __restrict__ eexp, float* __restrict__ den, int Ee) {
  int e = blockIdx.x * blockDim.x + threadIdx.x;
  if (e >= Ee) return;
  int s = (int)ei[e], d = (int)ei[Ee + e];
#pragma unroll
  for (int h = 0; h < NHEAD; ++h) {
    float l  = leaky(a_src[s * 4 + h] + a_dst[d * 4 + h]);
    float ex = expf(l - m[d * 4 + h]);
    eexp[e * 4 + h] = ex;
    atomicAdd(&den[d * 4 + h], ex);
  }
}

// ---------- edge pass 3: alpha + message scatter (wave32 per edge) ----------
__global__ __launch_bounds__(256) void k_scatter(const long long* __restrict__ ei,
                                                 const float* __restrict__ xp,
                                                 const float* __restrict__ eexp,
                                                 const float* __restrict__ den,
                                                 float* __restrict__ acc,
                                                 float* __restrict__ alpha_out, int Ee) {
  int e    = (blockIdx.x * blockDim.x + threadIdx.x) >> 5;
  int lane = threadIdx.x & 31;
  if (e >= Ee) return;
  int s = (int)ei[e], d = (int)ei[Ee + e];
  float al[NHEAD];
#pragma unroll
  for (int h = 0; h < NHEAD; ++h) al[h] = eexp[e * 4 + h] / (den[d * 4 + h] + 1e-16f);
  if (lane < NHEAD) alpha_out[(size_t)e * 4 + lane] = al[lane];
  const float* xs = xp  + (size_t)s * HC;
  float*       ao = acc + (size_t)d * HC;
#pragma unroll
  for (int i = 0; i < 8; ++i) {                 // channel c = lane + 32*i, head = i/2
    int c = lane + 32 * i;
    atomicAdd(&ao[c], xs[c] * al[i >> 1]);
  }
}

// ---------- BN1 stats (x1 = acc + bias) ----------
__global__ __launch_bounds__(256) void k_stats1(const float* __restrict__ acc,
                                                const float* __restrict__ bias,
                                                float* __restrict__ stats, int Nn) {
  int t = threadIdx.x;
  float s = 0.0f, s2 = 0.0f;
  for (int r = blockIdx.x; r < Nn; r += gridDim.x) {
    float v = acc[(size_t)r * HC + t] + bias[t];
    s += v; s2 += v * v;
  }
  atomicAdd(&stats[t], s);
  atomicAdd(&stats[256 + t], s2);
}

__global__ void k_fin1(float* __restrict__ stats, const float* __restrict__ gamma,
                       const float* __restrict__ beta, int Nn) {
  int t = threadIdx.x;
  float inv = 1.0f / (float)Nn;
  float mu  = stats[t] * inv;
  float var = stats[256 + t] * inv - mu * mu;
  var = var < 0.0f ? 0.0f : var;
  float sc = gamma[t] * rsqrtf(var + BN_EPS);
  stats[1024 + t] = sc;
  stats[1280 + t] = beta[t] - mu * sc;
}

// ---------- apply BN1 + ELU, store y, accumulate BN2 stats ----------
__global__ __launch_bounds__(256) void k_bn1_elu(const float* __restrict__ acc,
                                                 const float* __restrict__ bias,
                                                 const float* __restrict__ stats,
                                                 float* __restrict__ y,
                                                 float* __restrict__ stats_rw, int Nn) {
  int t = threadIdx.x;
  float sc = stats[1024 + t], sh = stats[1280 + t];
  float s = 0.0f, s2 = 0.0f;
  for (int r = blockIdx.x; r < Nn; r += gridDim.x) {
    float x1 = acc[(size_t)r * HC + t] + bias[t];
    float x2 = x1 * sc + sh;
    float v  = x2 > 0.0f ? x2 : expm1f(x2);   // ELU(alpha=1)
    y[(size_t)r * HC + t] = v;
    s += v; s2 += v * v;
  }
  atomicAdd(&stats_rw[512 + t], s);
  atomicAdd(&stats_rw[768 + t], s2);
}

__global__ void k_fin2(float* __restrict__ stats, const float* __restrict__ gamma,
                       const float* __restrict__ beta, int Nn) {
  int t = threadIdx.x;
  float inv = 1.0f / (float)Nn;
  float mu  = stats[512 + t] * inv;
  float var = stats[768 + t] * inv - mu * mu;
  var = var < 0.0f ? 0.0f : var;
  float sc = gamma[t] * rsqrtf(var + BN_EPS);
  stats[1536 + t] = sc;
  stats[1792 + t] = beta[t] - mu * sc;
}

// ---------- apply BN2, write heads layout [H,N,C] ----------
__global__ __launch_bounds__(256) void k_out(const float* __restrict__ y,
                                             const float* __restrict__ stats,
                                             float* __restrict__ out, int Nn) {
  int t = threadIdx.x;
  float sc = stats[1536 + t], sh = stats[1792 + t];
  int h = t >> 6, c = t & 63;
  for (int r = blockIdx.x; r < Nn; r += gridDim.x) {
    float v = y[(size_t)r * HC + t] * sc + sh;
    out[(size_t)h * Nn * CPH + (size_t)r * CPH + c] = v;
  }
}

extern "C" void kernel_launch(void* const* d_in, const int* in_sizes, int n_in,
                              void* d_out, int out_size, void* d_ws, size_t ws_size,
                              hipStream_t stream) {
  const float*     x       = (const float*)d_in[0];
  const long long* ei      = (const long long*)d_in[1];   // int64 edge_index [2,E]
  const float*     W       = (const float*)d_in[2];
  const float*     att_src = (const float*)d_in[3];
  const float*     att_dst = (const float*)d_in[4];
  const float*     bias    = (const float*)d_in[5];
  const float*     gamma   = (const float*)d_in[6];
  const float*     beta    = (const float*)d_in[7];

  const int Nn = in_sizes[0] / IN_DIM;   // 10000
  const int Ee = in_sizes[1] / 2;        // 320000
  const int nnh = Nn * NHEAD;

  // workspace layout (floats)
  float*    ws    = (float*)d_ws;
  float*    xp    = ws;                         // Nn*256  (reused as y after scatter)
  float*    acc   = xp + (size_t)Nn * HC;       // Nn*256
  float*    a_src = acc + (size_t)Nn * HC;      // Nn*4
  float*    a_dst = a_src + nnh;                // Nn*4
  unsigned* mbits = (unsigned*)(a_dst + nnh);   // Nn*4
  float*    m     = (float*)mbits + nnh;        // Nn*4
  float*    den   = m + nnh;                    // Nn*4
  float*    eexp  = den + nnh;                  // Ee*4
  float*    stats = eexp + (size_t)Ee * 4;      // 2048

  float* out_heads = (float*)d_out;                          // [H,Nn,64]
  float* out_alpha = out_heads + (size_t)NHEAD * Nn * CPH;   // [Ee,4]

  const int nacc = Nn * HC;
  k_init<<<(nacc + 255) / 256, 256, 0, stream>>>(acc, den, mbits, stats, nacc, nnh);

  k_gemm<<<(Nn + 15) / 16, 128, 0, stream>>>(x, W, xp, Nn);

  k_attn<<<(nnh + 255) / 256, 256, 0, stream>>>(xp, att_src, att_dst, a_src, a_dst, nnh);

  k_edge_max<<<(Ee + 255) / 256, 256, 0, stream>>>(ei, a_src, a_dst, mbits, Ee);
  k_node_m<<<(nnh + 255) / 256, 256, 0, stream>>>(mbits, m, nnh);
  k_edge_exp<<<(Ee + 255) / 256, 256, 0, stream>>>(ei, a_src, a_dst, m, eexp, den, Ee);
  k_scatter<<<(Ee * 32 + 255) / 256, 256, 0, stream>>>(ei, xp, eexp, den, acc, out_alpha, Ee);

  k_stats1<<<240, 256, 0, stream>>>(acc, bias, stats, Nn);
  k_fin1<<<1, 256, 0, stream>>>(stats, gamma, beta, Nn);
  k_bn1_elu<<<240, 256, 0, stream>>>(acc, bias, stats, /*y=*/xp, stats, Nn);
  k_fin2<<<1, 256, 0, stream>>>(stats, gamma, beta, Nn);
  k_out<<<240, 256, 0, stream>>>(/*y=*/xp, stats, out_heads, Nn);
}